// GATNet_14293651161150
// MI455X (gfx1250) — compile-verified
//
#include <hip/hip_runtime.h>
#include <cstdint>

#define NN      50000
#define NE      1600000
#define ET      (NE + NN)     // edges + self loops
#define IN_C    256
#define HID_C   128
#define HEADS   8
#define OUT_C   16
#define SLOPE   0.2f

typedef float v2f __attribute__((ext_vector_type(2)));
typedef float v8f __attribute__((ext_vector_type(8)));

// ---- monotone float<->uint encoding so atomicMax(u32) == float max ----
__device__ __forceinline__ unsigned enc_f32(float f) {
    unsigned u = __float_as_uint(f);
    return (u & 0x80000000u) ? ~u : (u | 0x80000000u);
}
__device__ __forceinline__ float dec_f32(unsigned e) {
    unsigned u = (e & 0x80000000u) ? (e & 0x7FFFFFFFu) : ~e;
    return __uint_as_float(u);
}

__global__ void k_zero(unsigned* __restrict__ p, long n) {
    long i = (long)blockIdx.x * 256 + threadIdx.x;
    if (i < n) p[i] = 0u;
}

// ---------------- GEMM1: h1[NN,128] = x[NN,256] @ W1[256,128] ----------------
// grid = NN/16 blocks, 256 thr (8 waves). Wave w -> columns [16w,16w+16).
__global__ void k_gemm1(const float* __restrict__ X, const float* __restrict__ W,
                        float* __restrict__ H) {
    const int wave = threadIdx.x >> 5;
    const int lane = threadIdx.x & 31;
    const int mBase = blockIdx.x << 4;
    const int nBase = wave << 4;
    const int lm = lane & 15;          // A row / B,C col within tile
    const int kh = (lane >> 4) << 1;   // K sub-offset for this lane
    const float* xr = X + (size_t)(mBase + lm) * IN_C + kh;
    const float* wp = W + (size_t)kh * HID_C + nBase + lm;
    v8f acc = {};
    #pragma unroll 8
    for (int kb = 0; kb < IN_C; kb += 4) {
        v2f a = *(const v2f*)(xr + kb);
        v2f b;
        b.x = wp[(size_t)kb * HID_C];
        b.y = wp[(size_t)kb * HID_C + HID_C];
        acc = __builtin_amdgcn_wmma_f32_16x16x4_f32(false, a, false, b,
                                                    (short)0, acc, false, false);
    }
    // C layout: vgpr r -> row m = r + 8*(lane>>4), col n = lane&15
    float* hp = H + (size_t)(mBase + ((lane >> 4) << 3)) * HID_C + nBase + lm;
    #pragma unroll
    for (int r = 0; r < 8; ++r) hp[(size_t)r * HID_C] = acc[r];
}

// ---------------- GEMM2: h2[NN,16] = out1[NN,128] @ W2[128,16] ---------------
// grid = ceil(NN/128), 256 thr; wave w -> rows [blk*128 + 16w, +16), all 16 cols.
__global__ void k_gemm2(const float* __restrict__ X, const float* __restrict__ W,
                        float* __restrict__ H) {
    const int wave = threadIdx.x >> 5;
    const int lane = threadIdx.x & 31;
    const int mBase = (blockIdx.x << 7) + (wave << 4);
    if (mBase >= NN) return;           // NN is a multiple of 16, tiles are whole
    const int lm = lane & 15;
    const int kh = (lane >> 4) << 1;
    const float* xr = X + (size_t)(mBase + lm) * HID_C + kh;
    const float* wp = W + (size_t)kh * OUT_C + lm;
    v8f acc = {};
    #pragma unroll 8
    for (int kb = 0; kb < HID_C; kb += 4) {
        v2f a = *(const v2f*)(xr + kb);
        v2f b;
        b.x = wp[(size_t)kb * OUT_C];
        b.y = wp[(size_t)kb * OUT_C + OUT_C];
        acc = __builtin_amdgcn_wmma_f32_16x16x4_f32(false, a, false, b,
                                                    (short)0, acc, false, false);
    }
    float* hp = H + (size_t)(mBase + ((lane >> 4) << 3)) * OUT_C + lm;
    #pragma unroll
    for (int r = 0; r < 8; ++r) hp[(size_t)r * OUT_C] = acc[r];
}

// -------- attention scalars: a_src[n,h] = <h[n,h,:], att_src[h,:]> (C=16) ----
__global__ void k_attvec(const float* __restrict__ H, const float* __restrict__ attS,
                         const float* __restrict__ attD, float* __restrict__ aS,
                         float* __restrict__ aD, int nH) {
    int idx = blockIdx.x * blockDim.x + threadIdx.x;
    if (idx >= NN * nH) return;
    int hd = idx % nH;
    const float4* hp = (const float4*)(H + (size_t)idx * 16);
    const float4* sp = (const float4*)(attS + hd * 16);
    const float4* dp = (const float4*)(attD + hd * 16);
    float s = 0.f, d = 0.f;
    #pragma unroll
    for (int j = 0; j < 4; ++j) {
        float4 h4 = hp[j], s4 = sp[j], d4 = dp[j];
        s += h4.x * s4.x + h4.y * s4.y + h4.z * s4.z + h4.w * s4.w;
        d += h4.x * d4.x + h4.y * d4.y + h4.z * d4.z + h4.w * d4.w;
    }
    aS[idx] = s; aD[idx] = d;
}

// -------- edge pass 1: segment max of leaky_relu(a_src[s]+a_dst[d]) ----------
template <int NH>
__global__ void k_edge_max(const int* __restrict__ esrc, const int* __restrict__ edst,
                           const float* __restrict__ aS, const float* __restrict__ aD,
                           unsigned* __restrict__ amax) {
    int e = blockIdx.x * blockDim.x + threadIdx.x;
    if (e >= ET) return;
    int s, d;
    if (e < NE) { s = esrc[e]; d = edst[e]; } else { s = e - NE; d = s; }
    #pragma unroll
    for (int hd = 0; hd < NH; ++hd) {
        float al = aS[(size_t)s * NH + hd] + aD[(size_t)d * NH + hd];
        al = al > 0.f ? al : al * SLOPE;
        atomicMax(&amax[(size_t)d * NH + hd], enc_f32(al));
    }
}

// -------- edge pass 2: w = exp(alpha - max); denom += w; accum += w*h[src] ---
// one thread per (edge, head), 16 channels per head
template <int NH>
__global__ void k_edge_accum(const int* __restrict__ esrc, const int* __restrict__ edst,
                             const float* __restrict__ aS, const float* __restrict__ aD,
                             const unsigned* __restrict__ amax, const float* __restrict__ H,
                             float* __restrict__ denom, float* __restrict__ accum) {
    long idx = (long)blockIdx.x * blockDim.x + threadIdx.x;
    if (idx >= (long)ET * NH) return;
    int e  = (int)(idx / NH);
    int hd = (int)(idx - (long)e * NH);
    int s, d;
    if (e < NE) { s = esrc[e]; d = edst[e]; } else { s = e - NE; d = s; }
    float al = aS[(size_t)s * NH + hd] + aD[(size_t)d * NH + hd];
    al = al > 0.f ? al : al * SLOPE;
    float w = __expf(al - dec_f32(amax[(size_t)d * NH + hd]));
    atomicAdd(&denom[(size_t)d * NH + hd], w);
    const float4* hp = (const float4*)(H + ((size_t)s * NH + hd) * 16);
    float* op = accum + ((size_t)d * NH + hd) * 16;
    #pragma unroll
    for (int j = 0; j < 4; ++j) {
        float4 v = hp[j];
        atomicAdd(op + 4 * j + 0, w * v.x);
        atomicAdd(op + 4 * j + 1, w * v.y);
        atomicAdd(op + 4 * j + 2, w * v.z);
        atomicAdd(op + 4 * j + 3, w * v.w);
    }
}

// -------- finalize: accum = (opt relu)(accum/denom + bias), in place ---------
__global__ void k_fin(float* __restrict__ accum, const float* __restrict__ denom,
                      const float* __restrict__ bias, int rowMask, long total, int relu) {
    long idx = (long)blockIdx.x * blockDim.x + threadIdx.x;
    if (idx >= total) return;
    int c = (int)(idx & rowMask);      // channel index within node row (pow2)
    long nh = idx >> 4;                // (n*nH + head)
    float v = accum[idx] / denom[nh] + bias[c];
    if (relu) v = fmaxf(v, 0.f);
    accum[idx] = v;
}

extern "C" void kernel_launch(void* const* d_in, const int* in_sizes, int n_in,
                              void* d_out, int out_size, void* d_ws, size_t ws_size,
                              hipStream_t stream) {
    const float* x     = (const float*)d_in[0];
    const int*   ei    = (const int*)  d_in[1];
    const float* W1    = (const float*)d_in[2];
    const float* attS1 = (const float*)d_in[3];
    const float* attD1 = (const float*)d_in[4];
    const float* bias1 = (const float*)d_in[5];
    const float* W2    = (const float*)d_in[6];
    const float* attS2 = (const float*)d_in[7];
    const float* attD2 = (const float*)d_in[8];
    const float* bias2 = (const float*)d_in[9];
    float* out = (float*)d_out;
    const int* esrc = ei;
    const int* edst = ei + NE;

    // ---- workspace layout (zero-init zone first, contiguous) ----
    unsigned* amax1  = (unsigned*)d_ws;                       // NN*8
    float*    denom1 = (float*)(amax1 + (size_t)NN * 8);      // NN*8
    float*    accum1 = denom1 + (size_t)NN * 8;               // NN*128 (-> out1 in place)
    unsigned* amax2  = (unsigned*)(accum1 + (size_t)NN * 128);// NN
    float*    denom2 = (float*)(amax2 + NN);                  // NN
    // non-zeroed scratch
    float* h1  = denom2 + NN;                                 // NN*128
    float* aS1 = h1 + (size_t)NN * 128;                       // NN*8
    float* aD1 = aS1 + (size_t)NN * 8;                        // NN*8
    float* h2  = aD1 + (size_t)NN * 8;                        // NN*16
    float* aS2 = h2 + (size_t)NN * 16;                        // NN
    float* aD2 = aS2 + NN;                                    // NN

    const long znWs  = (long)NN * (8 + 8 + 128 + 1 + 1);
    const long znOut = (long)NN * OUT_C;
    k_zero<<<(int)((znWs  + 255) / 256), 256, 0, stream>>>((unsigned*)d_ws, znWs);
    k_zero<<<(int)((znOut + 255) / 256), 256, 0, stream>>>((unsigned*)out, znOut);

    // ---------------- layer 1 (HEADS=8, C=16) ----------------
    k_gemm1<<<NN / 16, 256, 0, stream>>>(x, W1, h1);
    k_attvec<<<(NN * HEADS + 255) / 256, 256, 0, stream>>>(h1, attS1, attD1, aS1, aD1, HEADS);
    k_edge_max<HEADS><<<(ET + 255) / 256, 256, 0, stream>>>(esrc, edst, aS1, aD1, amax1);
    {
        long n = (long)ET * HEADS;
        k_edge_accum<HEADS><<<(int)((n + 255) / 256), 256, 0, stream>>>(
            esrc, edst, aS1, aD1, amax1, h1, denom1, accum1);
    }
    {
        long n = (long)NN * HID_C;
        k_fin<<<(int)((n + 255) / 256), 256, 0, stream>>>(accum1, denom1, bias1,
                                                          HID_C - 1, n, 1);
    }

    // ---------------- layer 2 (HEADS=1, C=16) ----------------
    k_gemm2<<<(NN + 127) / 128, 256, 0, stream>>>(accum1, W2, h2);
    k_attvec<<<(NN + 255) / 256, 256, 0, stream>>>(h2, attS2, attD2, aS2, aD2, 1);
    k_edge_max<1><<<(ET + 255) / 256, 256, 0, stream>>>(esrc, edst, aS2, aD2, amax2);
    {
        long n = (long)ET;
        k_edge_accum<1><<<(int)((n + 255) / 256), 256, 0, stream>>>(
            esrc, edst, aS2, aD2, amax2, h2, denom2, out);
    }
    {
        long n = (long)NN * OUT_C;
        k_fin<<<(int)((n + 255) / 256), 256, 0, stream>>>(out, denom2, bias2,
                                                          OUT_C - 1, n, 0);
    }
}